// EdgeAttnGATConv_11570641895627
// MI455X (gfx1250) — compile-verified
//
#include <hip/hip_runtime.h>

#define N_NODES 50000
#define N_EDGES 800000
#define IN_CH   128
#define HEADS   4
#define OUT_CH  32
#define HC      128      // HEADS * OUT_CH
#define NEG_SLOPE 0.2f

typedef float v2f __attribute__((ext_vector_type(2)));
typedef float v8f __attribute__((ext_vector_type(8)));

// ---------------------------------------------------------------------------
// Kernel 0: out = bias (broadcast), seg_max = 0 (matches torch scatter-amax
// starting from zeros), seg_sum = 0.
// ---------------------------------------------------------------------------
__global__ void init_kernel(float* __restrict__ out, const float* __restrict__ bias,
                            float* __restrict__ seg_max, float* __restrict__ seg_sum) {
  int idx = blockIdx.x * blockDim.x + threadIdx.x;
  if (idx < N_NODES * HC) out[idx] = bias[idx & (HC - 1)];
  if (idx < N_NODES * HEADS) { seg_max[idx] = 0.0f; seg_sum[idx] = 0.0f; }
}

// ---------------------------------------------------------------------------
// Kernel 1: fp32 WMMA projections  x_src = x @ W_src, x_dst = x @ W_dst.
// One wave per 16x16 output tile; K=128 swept as 32 chained
// v_wmma_f32_16x16x4_f32 ops on the f32 accumulator.
// A layout (16x4 f32): lanes 0-15 hold K=0,1  lanes 16-31 hold K=2,3.
// B layout (4x16 f32): VGPR0 = rows K={0|2}, VGPR1 = rows K={1|3}.
// C/D layout (16x16 f32): VGPR j = row (j + 8*(lane>>4)), col = lane&15.
// ---------------------------------------------------------------------------
__global__ void proj_wmma_kernel(const float* __restrict__ x,
                                 const float* __restrict__ W_src,
                                 const float* __restrict__ W_dst,
                                 float* __restrict__ x_src,
                                 float* __restrict__ x_dst) {
  const int lane = threadIdx.x;                 // 0..31, wave32
  const int row0 = blockIdx.x * 16;
  const int col0 = blockIdx.y * 16;
  const float* __restrict__ W = blockIdx.z ? W_dst : W_src;
  float* __restrict__       Y = blockIdx.z ? x_dst : x_src;

  const int mn   = lane & 15;
  const int koff = (lane >> 4) << 1;            // 0 or 2

  const float* __restrict__ arow = x + (size_t)(row0 + mn) * IN_CH + koff;
  const float* __restrict__ bcol = W + (size_t)koff * HC + col0 + mn;

  v8f acc = {0.f, 0.f, 0.f, 0.f, 0.f, 0.f, 0.f, 0.f};
#pragma unroll 4
  for (int kk = 0; kk < IN_CH / 4; ++kk) {
    v2f a, b;
    a.x = arow[kk * 4 + 0];                     // K = 4*kk + koff
    a.y = arow[kk * 4 + 1];                     // K = 4*kk + koff + 1
    b.x = bcol[(size_t)(kk * 4) * HC];          // row K = 4*kk + koff
    b.y = bcol[(size_t)(kk * 4 + 1) * HC];      // row K = 4*kk + koff + 1
    acc = __builtin_amdgcn_wmma_f32_16x16x4_f32(
        /*neg_a=*/false, a, /*neg_b=*/false, b,
        /*c_mod=*/(short)0, acc, /*reuse_a=*/false, /*reuse_b=*/false);
  }

  const int mrow = (lane >> 4) * 8;
#pragma unroll
  for (int j = 0; j < 8; ++j)
    Y[(size_t)(row0 + mrow + j) * HC + col0 + mn] = acc[j];
}

// ---------------------------------------------------------------------------
// Kernel 2: per-edge attention logits + segment max.
// One wave per edge iteration; 32 lanes = channel dim C, loop over 4 heads.
// Edge-feature GEMM (K=2) folded in as two FMAs per lane.
// seg_max via int-bitpattern atomicMax: init is 0.0f and the reference clamps
// at 0, so only positive floats (order-isomorphic to ints) can win.
// ---------------------------------------------------------------------------
__global__ void alpha_max_kernel(const float* __restrict__ x_src,
                                 const float* __restrict__ x_dst,
                                 const int*   __restrict__ edge_index,
                                 const float* __restrict__ edge_attr,
                                 const float* __restrict__ W_edge,
                                 const float* __restrict__ att,
                                 float* __restrict__ alpha,
                                 float* __restrict__ seg_max) {
  const int lane   = threadIdx.x & 31;
  const int wave   = (blockIdx.x * blockDim.x + threadIdx.x) >> 5;
  const int nwaves = (gridDim.x * blockDim.x) >> 5;

  for (int e = wave; e < N_EDGES; e += nwaves) {
    const int row = edge_index[e];
    const int col = edge_index[N_EDGES + e];
    const float ea0 = edge_attr[2 * e + 0];
    const float ea1 = edge_attr[2 * e + 1];
    const float* __restrict__ xs = x_src + (size_t)row * HC;
    const float* __restrict__ xd = x_dst + (size_t)col * HC;
#pragma unroll
    for (int h = 0; h < HEADS; ++h) {
      const int c = h * OUT_CH + lane;
      float v = xs[c] + xd[c] + ea0 * W_edge[c] + ea1 * W_edge[HC + c];
      v = (v > 0.f) ? v : NEG_SLOPE * v;        // leaky_relu
      float p = v * att[c];
#pragma unroll
      for (int off = 16; off > 0; off >>= 1)    // wave32 reduction
        p += __shfl_xor(p, off, 32);
      if (lane == 0) {
        alpha[(size_t)e * HEADS + h] = p;
        if (p > 0.f)
          atomicMax((int*)seg_max + (size_t)col * HEADS + h, __float_as_int(p));
      }
    }
  }
}

// ---------------------------------------------------------------------------
// Kernel 3: alpha = exp(alpha - seg_max[col]); seg_sum += alpha.
// ---------------------------------------------------------------------------
__global__ void exp_sum_kernel(const int*   __restrict__ edge_index,
                               float* __restrict__ alpha,
                               const float* __restrict__ seg_max,
                               float* __restrict__ seg_sum) {
  int idx = blockIdx.x * blockDim.x + threadIdx.x;
  if (idx >= N_EDGES * HEADS) return;
  int e = idx >> 2;                              // / HEADS
  int h = idx & 3;                               // % HEADS
  int col = edge_index[N_EDGES + e];
  float a = expf(alpha[idx] - seg_max[(size_t)col * HEADS + h]);
  alpha[idx] = a;
  atomicAdd(seg_sum + (size_t)col * HEADS + h, a);
}

// ---------------------------------------------------------------------------
// Kernel 4: out[col] += (alpha / (seg_sum[col] + 1e-16)) * x_dst[col].
// Wave per edge, 32 lanes = channel dim -> coalesced 128B atomic lines.
// ---------------------------------------------------------------------------
__global__ void scatter_kernel(const float* __restrict__ x_dst,
                               const int*   __restrict__ edge_index,
                               const float* __restrict__ alpha,
                               const float* __restrict__ seg_sum,
                               float* __restrict__ out) {
  const int lane   = threadIdx.x & 31;
  const int wave   = (blockIdx.x * blockDim.x + threadIdx.x) >> 5;
  const int nwaves = (gridDim.x * blockDim.x) >> 5;

  for (int e = wave; e < N_EDGES; e += nwaves) {
    const int col = edge_index[N_EDGES + e];
    const float* __restrict__ xd = x_dst + (size_t)col * HC;
    float* __restrict__ o = out + (size_t)col * HC;
#pragma unroll
    for (int h = 0; h < HEADS; ++h) {
      float w = alpha[(size_t)e * HEADS + h] /
                (seg_sum[(size_t)col * HEADS + h] + 1e-16f);
      atomicAdd(o + h * OUT_CH + lane, xd[h * OUT_CH + lane] * w);
    }
  }
}

// ---------------------------------------------------------------------------
extern "C" void kernel_launch(void* const* d_in, const int* in_sizes, int n_in,
                              void* d_out, int out_size, void* d_ws, size_t ws_size,
                              hipStream_t stream) {
  const float* x          = (const float*)d_in[0];
  const int*   edge_index = (const int*)  d_in[1];
  const float* edge_attr  = (const float*)d_in[2];
  const float* W_src      = (const float*)d_in[3];
  const float* W_dst      = (const float*)d_in[4];
  const float* W_edge     = (const float*)d_in[5];
  const float* att        = (const float*)d_in[6];
  const float* bias       = (const float*)d_in[7];
  float* out = (float*)d_out;

  // Workspace layout (floats): 65.6 MB total.
  float* ws      = (float*)d_ws;
  float* x_src   = ws;                               // N*128
  float* x_dst   = x_src + (size_t)N_NODES * HC;     // N*128
  float* alpha   = x_dst + (size_t)N_NODES * HC;     // E*H
  float* seg_max = alpha + (size_t)N_EDGES * HEADS;  // N*H
  float* seg_sum = seg_max + (size_t)N_NODES * HEADS;// N*H

  init_kernel<<<(N_NODES * HC + 255) / 256, 256, 0, stream>>>(out, bias, seg_max, seg_sum);

  dim3 pgrid(N_NODES / 16, HC / 16, 2);              // 3125 x 8 x 2 tiles
  proj_wmma_kernel<<<pgrid, 32, 0, stream>>>(x, W_src, W_dst, x_src, x_dst);

  alpha_max_kernel<<<2048, 256, 0, stream>>>(x_src, x_dst, edge_index, edge_attr,
                                             W_edge, att, alpha, seg_max);

  exp_sum_kernel<<<(N_EDGES * HEADS + 255) / 256, 256, 0, stream>>>(edge_index, alpha,
                                                                    seg_max, seg_sum);

  scatter_kernel<<<2048, 256, 0, stream>>>(x_dst, edge_index, alpha, seg_sum, out);
}